// IndexNSW_65352222376251
// MI455X (gfx1250) — compile-verified
//
#include <hip/hip_runtime.h>
#include <stdint.h>

// Problem constants (from reference): N, D, B, R, EF, K
#define NN     500000
#define DIMV   128
#define BATCH  512
#define RD     32
#define EFS    64
#define KK     10
#define CAP    128            // 64 beam + 32 new + 32 pad slots
#define ROWP   132            // padded LDS row stride (dwords): 128 + 4 -> kills bank conflicts
#define INF_D  1e30f

typedef __attribute__((ext_vector_type(2)))  float    v2f;
typedef __attribute__((ext_vector_type(4)))  float    v4f;
typedef __attribute__((ext_vector_type(8)))  float    v8f;
typedef __attribute__((ext_vector_type(4)))  unsigned v4u;
typedef __attribute__((ext_vector_type(8)))  int      v8i;
typedef __attribute__((ext_vector_type(4)))  int      v4i;

#if defined(__has_builtin)
#  if __has_builtin(__builtin_amdgcn_wmma_f32_16x16x4_f32)
#    define HAVE_WMMA_F32X4 1
#  endif
#  if __has_builtin(__builtin_amdgcn_tensor_load_to_lds)
#    define HAVE_TDM 1
#  endif
#endif

// ---------------------------------------------------------------------------
// Issue the gather of 32 rows (ids in LDS s_nodes[0..31]) into s_vec (ROWP
// stride).  TDM path: 4 gather-mode descriptors, 8 x 32-bit row indices each,
// pad_interval=128 dwords / pad_amount=4 dwords -> rows land at 132-dword
// stride.  Tracked on TENSORcnt so the caller can overlap scalar work.
// ---------------------------------------------------------------------------
__device__ __forceinline__ void gather32_issue(const float* __restrict__ storage,
                                               const int* s_nodes,
                                               float (*s_vec)[ROWP],
                                               int lane)
{
#ifdef HAVE_TDM
    unsigned lds_base = (unsigned)(uintptr_t)(&s_vec[0][0]);
    unsigned long long ga = (unsigned long long)(uintptr_t)storage;
    #pragma unroll
    for (int g = 0; g < 4; ++g) {
        v4u g0; v8i g1; v4i g2, g3; v8i g23;
        // group0: count=1, gather_mode=1, 32-bit indices, lds_addr, global_addr, type=2
        g0[0] = 0xC0000001u;
        g0[1] = lds_base + (unsigned)(g * 8 * ROWP * 4);
        g0[2] = (unsigned)(ga & 0xFFFFFFFFu);
        g0[3] = (unsigned)((ga >> 32) & 0x1FFFFFFu) | 0x80000000u;
        // group1: data_size=4B, pad_enable, pad_interval=128 dw, pad_amount=4 dw,
        //         tensor_dim0=128, tensor_dim1=N, tile_dim0=128, tile_dim1=8 idx,
        //         tensor_dim0_stride=128
        g1[0] = (int)((2u << 16) | (1u << 20) | (6u << 22) | (3u << 25));
        g1[1] = (int)((unsigned)DIMV << 16);                      // dim0[15:0]
        g1[2] = (int)(((unsigned)NN & 0xFFFFu) << 16);            // dim1[15:0]
        g1[3] = (int)(((unsigned)NN >> 16) | ((unsigned)DIMV << 16)); // dim1 hi | tile_dim0
        g1[4] = 8;                                                // tile_dim1 = #indices
        g1[5] = DIMV;                                             // dim0_stride lo
        g1[6] = 0;
        g1[7] = 0;
        // groups 2-3: eight 32-bit row indices (force SGPR via readfirstlane)
        g2[0] = __builtin_amdgcn_readfirstlane(s_nodes[g * 8 + 0]);
        g2[1] = __builtin_amdgcn_readfirstlane(s_nodes[g * 8 + 1]);
        g2[2] = __builtin_amdgcn_readfirstlane(s_nodes[g * 8 + 2]);
        g2[3] = __builtin_amdgcn_readfirstlane(s_nodes[g * 8 + 3]);
        g3[0] = __builtin_amdgcn_readfirstlane(s_nodes[g * 8 + 4]);
        g3[1] = __builtin_amdgcn_readfirstlane(s_nodes[g * 8 + 5]);
        g3[2] = __builtin_amdgcn_readfirstlane(s_nodes[g * 8 + 6]);
        g3[3] = __builtin_amdgcn_readfirstlane(s_nodes[g * 8 + 7]);
        // 5th operand (uncharacterized on this toolchain): pass groups 2+3
        // concatenated so the indices are present under either interpretation.
        g23[0] = g2[0]; g23[1] = g2[1]; g23[2] = g2[2]; g23[3] = g2[3];
        g23[4] = g3[0]; g23[5] = g3[1]; g23[6] = g3[2]; g23[7] = g3[3];
        __builtin_amdgcn_tensor_load_to_lds(g0, g1, g2, g3, g23, 0);
    }
#else
    // Cooperative fallback: wave loads one 512B row per iteration (coalesced).
    for (int r = 0; r < RD; ++r) {
        int node = s_nodes[r];
        v4f v = *(const v4f*)(storage + (size_t)node * DIMV + lane * 4);
        *(v4f*)(&s_vec[r][lane * 4]) = v;
    }
#endif
}

__device__ __forceinline__ void gather32_wait()
{
#ifdef HAVE_TDM
    __builtin_amdgcn_s_wait_tensorcnt(0);
#endif
    __syncthreads();
}

// ---------------------------------------------------------------------------
// Distances of the 32 gathered rows vs the query.
// dist = ||x||^2 - 2 x.q + ||q||^2 ; the dot products go through the f32 WMMA
// (two 16x16 accumulators, K accumulated in chunks of 4), norms on the VALU.
// ---------------------------------------------------------------------------
__device__ __forceinline__ float dist32(const float (*s_vec)[ROWP],
                                        const float* s_q,
                                        float (*s_stage)[256],
                                        float qn, int lane)
{
    const int hf   = lane >> 4;   // 0/1 half of the wave
    const int lrow = lane & 15;

#ifdef HAVE_WMMA_F32X4
    v8f c0 = {}; v8f c1 = {};
    for (int k0 = 0; k0 < DIMV; k0 += 4) {
        // B (4x16): query replicated into all 16 columns
        v2f bq; bq.x = s_q[k0 + hf];       bq.y = s_q[k0 + 2 + hf];
        // A (16x4): rows 0..15 (c0) / 16..31 (c1)
        v2f a0; a0.x = s_vec[lrow][k0 + 2*hf];      a0.y = s_vec[lrow][k0 + 2*hf + 1];
        v2f a1; a1.x = s_vec[16 + lrow][k0 + 2*hf]; a1.y = s_vec[16 + lrow][k0 + 2*hf + 1];
        c0 = __builtin_amdgcn_wmma_f32_16x16x4_f32(false, a0, false, bq, (short)0, c0, false, false);
        c1 = __builtin_amdgcn_wmma_f32_16x16x4_f32(false, a1, false, bq, (short)0, c1, false, false);
    }
    // stage C tiles in LDS, then lane L extracts element (M = L%16, N = L%16)
    __syncthreads();
    #pragma unroll
    for (int i = 0; i < 8; ++i) {
        s_stage[0][lane * 8 + i] = c0[i];
        s_stage[1][lane * 8 + i] = c1[i];
    }
    __syncthreads();
    int m       = lrow;
    int srcLane = (m < 8) ? m : (m + 16);  // C layout: rows 0..7 lanes 0-15, 8..15 lanes 16-31
    int vg      = m & 7;
    float dot   = s_stage[hf][srcLane * 8 + vg];

    float nrm = 0.f;
    for (int c = 0; c < DIMV; c += 4) {
        v4f v = *(const v4f*)(&s_vec[lane][c]);
        nrm += v.x * v.x + v.y * v.y + v.z * v.z + v.w * v.w;
    }
    return nrm - 2.0f * dot + qn;
#else
    float acc = 0.f;
    for (int c = 0; c < DIMV; ++c) {
        float d = s_vec[lane][c] - s_q[c];
        acc += d * d;
    }
    return acc;
#endif
}

// ---------------------------------------------------------------------------
// Wave-cooperative bitonic sort of the 128-slot candidate array by distance.
// ---------------------------------------------------------------------------
__device__ __forceinline__ void sort128(int* s_id, float* s_d, int* s_ex, int lane)
{
    __syncthreads();
    for (int ksz = 2; ksz <= CAP; ksz <<= 1) {
        for (int j = ksz >> 1; j > 0; j >>= 1) {
            #pragma unroll
            for (int t = 0; t < CAP / 32; ++t) {
                int i = lane + 32 * t;
                int p = i ^ j;
                if (p > i) {
                    bool up = ((i & ksz) == 0);
                    float di = s_d[i], dp = s_d[p];
                    if (up ? (di > dp) : (di < dp)) {
                        s_d[i] = dp;  s_d[p] = di;
                        int ti = s_id[i]; s_id[i] = s_id[p]; s_id[p] = ti;
                        int te = s_ex[i]; s_ex[i] = s_ex[p]; s_ex[p] = te;
                    }
                }
            }
            __syncthreads();
        }
    }
}

// ---------------------------------------------------------------------------
// One wave32 per query: best-first NSW beam search, EF expansion steps.
// ---------------------------------------------------------------------------
__global__ __launch_bounds__(32) void nsw_search_kernel(
    const float* __restrict__ query,
    const float* __restrict__ storage,
    const int*   __restrict__ indices,
    const int*   __restrict__ entry,
    float*       __restrict__ out_ids,
    float*       __restrict__ out_dist)
{
    __shared__ float s_q[DIMV];
    __shared__ float s_vec[RD][ROWP];     // gathered rows (padded stride)
    __shared__ float s_stage[2][256];     // WMMA C-tile staging
    __shared__ int   s_id[CAP];
    __shared__ float s_d[CAP];
    __shared__ int   s_ex[CAP];
    __shared__ int   s_nid[RD];

    const int b    = blockIdx.x;
    const int lane = threadIdx.x;

    // query -> LDS, ||q||^2 via wave reduction
    v4f q4 = *(const v4f*)(query + (size_t)b * DIMV + lane * 4);
    *(v4f*)(s_q + lane * 4) = q4;
    float qn = q4.x * q4.x + q4.y * q4.y + q4.z * q4.z + q4.w * q4.w;
    for (int off = 16; off; off >>= 1) qn += __shfl_xor(qn, off, 32);

    // ---- initial beam: 64 entry points -------------------------------------
    int e0 = entry[b * EFS + lane];
    int e1 = entry[b * EFS + 32 + lane];
    s_id[lane]      = e0;  s_d[lane]      = INF_D; s_ex[lane]      = 0;
    s_id[32 + lane] = e1;  s_d[32 + lane] = INF_D; s_ex[32 + lane] = 0;
    s_id[64 + lane] = -1;  s_d[64 + lane] = INF_D; s_ex[64 + lane] = 1;
    s_id[96 + lane] = -1;  s_d[96 + lane] = INF_D; s_ex[96 + lane] = 1;
    __syncthreads();

    for (int g = 0; g < 2; ++g) {
        s_nid[lane] = s_id[g * 32 + lane];
        __syncthreads();
        gather32_issue(storage, s_nid, s_vec, lane);
        gather32_wait();
        float dd = dist32(s_vec, s_q, s_stage, qn, lane);
        s_d[g * 32 + lane] = dd;
        __syncthreads();
    }
    // dedup entry points: kill later copies (stable-first semantics)
    {
        int id0 = s_id[lane], id1 = s_id[32 + lane];
        int dead0 = 0, dead1 = 0;
        for (int j = 0; j < lane; ++j)      dead0 |= (s_id[j] == id0);
        for (int j = 0; j < 32 + lane; ++j) dead1 |= (s_id[j] == id1);
        __syncthreads();
        if (dead0) { s_d[lane] = INF_D;      s_ex[lane] = 1; }
        if (dead1) { s_d[32 + lane] = INF_D; s_ex[32 + lane] = 1; }
    }
    sort128(s_id, s_d, s_ex, lane);

    // ---- EF expansion steps ------------------------------------------------
    for (int step = 0; step < EFS; ++step) {
        // argmin over unexpanded beam slots (lexicographic (dist, idx) reduce)
        float v0 = s_ex[lane]      ? INF_D : s_d[lane];
        float v1 = s_ex[32 + lane] ? INF_D : s_d[32 + lane];
        float v  = v0; int idx = lane;
        if (v1 < v) { v = v1; idx = 32 + lane; }
        for (int off = 16; off; off >>= 1) {
            float ov = __shfl_xor(v, off, 32);
            int   oi = __shfl_xor(idx, off, 32);
            if (ov < v || (ov == v && oi < idx)) { v = ov; idx = oi; }
        }
        int sel   = idx;              // uniform across the wave
        int selid = s_id[sel];
        __syncthreads();
        if (lane == 0) s_ex[sel] = 1;

        // neighbor ids of the expanded node
        int nbr = indices[(size_t)selid * RD + lane];
        s_nid[lane] = nbr;
        __syncthreads();

        // kick off the 16KB row gather, overlap dedup with it
        gather32_issue(storage, s_nid, s_vec, lane);
        int dead = 0;
        for (int j = 0; j < EFS; ++j)  dead |= (s_id[j] == nbr);   // vs beam
        for (int j = 0; j < lane; ++j) dead |= (s_nid[j] == nbr);  // vs earlier lanes
        gather32_wait();

        float dd = dist32(s_vec, s_q, s_stage, qn, lane);

        // append 32 candidates + refresh INF pads, then re-select top-64
        s_id[64 + lane] = nbr;
        s_d[64 + lane]  = dead ? INF_D : dd;
        s_ex[64 + lane] = dead ? 1 : 0;
        s_id[96 + lane] = -1;
        s_d[96 + lane]  = INF_D;
        s_ex[96 + lane] = 1;
        sort128(s_id, s_d, s_ex, lane);
    }

    // ---- output top-K (beam already sorted ascending) ----------------------
    if (lane < KK) {
        out_ids[b * KK + lane]  = (float)s_id[lane];
        out_dist[b * KK + lane] = s_d[lane];
    }
}

extern "C" void kernel_launch(void* const* d_in, const int* in_sizes, int n_in,
                              void* d_out, int out_size, void* d_ws, size_t ws_size,
                              hipStream_t stream)
{
    const float* query   = (const float*)d_in[0];
    const float* storage = (const float*)d_in[1];
    const int*   indices = (const int*)d_in[2];
    const int*   entry   = (const int*)d_in[3];
    (void)in_sizes; (void)n_in; (void)d_ws; (void)ws_size; (void)out_size;

    float* out_ids  = (float*)d_out;               // ids[:, :K] (as float)
    float* out_dist = (float*)d_out + BATCH * KK;  // dists[:, :K]

    nsw_search_kernel<<<BATCH, 32, 0, stream>>>(query, storage, indices, entry,
                                                out_ids, out_dist);
}